// GNN_73727408603845
// MI455X (gfx1250) — compile-verified
//
#include <hip/hip_runtime.h>

#define N_NODES   100000
#define N_EDGES   1600000
#define E_TOT     (N_EDGES + N_NODES)
#define DIM       32
#define N_LAYERS  3
#define BN_EPS    1e-5f
#define AGG_BLOCKS 512
#define BLOCK     256
#define WPB       (BLOCK / 32)

typedef float v2f __attribute__((ext_vector_type(2)));
typedef float v8f __attribute__((ext_vector_type(8)));

// ---- degree / norm -------------------------------------------------------

__global__ void k_init_cnt(int* __restrict__ cnt) {
    int i = blockIdx.x * blockDim.x + threadIdx.x;
    if (i < N_NODES) cnt[i] = 1;                 // self loop
}

__global__ void k_count(const int* __restrict__ dst, int* __restrict__ cnt) {
    int e = blockIdx.x * blockDim.x + threadIdx.x;
    if (e < N_EDGES) atomicAdd(&cnt[dst[e]], 1);
}

__global__ void k_dinv(const int* __restrict__ cnt, float* __restrict__ dinv) {
    int i = blockIdx.x * blockDim.x + threadIdx.x;
    if (i < N_NODES) dinv[i] = rsqrtf((float)cnt[i]);   // cnt >= 1 always
}

// ---- exclusive scan of in-degrees (single 1024-thread block) -------------

__global__ void k_scan(const int* __restrict__ cnt, int* __restrict__ rowptr,
                       int* __restrict__ cursor) {
    const int SEG = (N_NODES + 1023) / 1024;     // 98
    __shared__ int sc[1024];
    int t   = threadIdx.x;
    int beg = t * SEG;
    int end = beg + SEG; if (end > N_NODES) end = N_NODES;
    int local = 0;
    for (int i = beg; i < end; ++i) local += cnt[i];
    sc[t] = local;
    __syncthreads();
    for (int off = 1; off < 1024; off <<= 1) {
        int v = (t >= off) ? sc[t - off] : 0;
        __syncthreads();
        sc[t] += v;
        __syncthreads();
    }
    int run = sc[t] - local;                     // exclusive base
    for (int i = beg; i < end; ++i) {
        rowptr[i] = run; cursor[i] = run;
        run += cnt[i];
    }
    if (t == 1023) rowptr[N_NODES] = sc[1023];   // total = E_TOT
}

// ---- CSR fill (edges + self loops), weights precomputed ------------------

__global__ void k_fill(const int* __restrict__ src, const int* __restrict__ dst,
                       const float* __restrict__ dinv, int* __restrict__ cursor,
                       int2* __restrict__ csr) {
    int i = blockIdx.x * blockDim.x + threadIdx.x;
    if (i >= E_TOT) return;
    int s, d;
    if (i < N_EDGES) { s = src[i]; d = dst[i]; }
    else             { s = i - N_EDGES; d = s; }           // self loop
    float w = dinv[s] * dinv[d];
    int slot = atomicAdd(&cursor[d], 1);
    csr[slot] = make_int2(s, __float_as_int(w));
}

// ---- pull-gather aggregation + relu + BN partial stats -------------------

__global__ __launch_bounds__(BLOCK)
void k_agg(const float* __restrict__ zin, float* __restrict__ zout,
           const int* __restrict__ rowptr, const int2* __restrict__ csr,
           float* __restrict__ partials) {
    const int lane = threadIdx.x & 31;           // lane == feature
    const int wave = threadIdx.x >> 5;
    const int gw   = blockIdx.x * WPB + wave;
    const int nw   = AGG_BLOCKS * WPB;
    float s1 = 0.f, s2 = 0.f;
    for (int node = gw; node < N_NODES; node += nw) {
        int j = rowptr[node], end = rowptr[node + 1];
        float acc = 0.f;
        for (; j < end; ++j) {
            __builtin_prefetch(&csr[j + 16], 0, 0);   // global_prefetch_b8 on CSR stream
            int2 e = csr[j];
            acc = fmaf(__int_as_float(e.y), zin[(size_t)e.x * DIM + lane], acc);
        }
        acc = fmaxf(acc, 0.f);                   // relu (BN input)
        zout[(size_t)node * DIM + lane] = acc;
        s1 += acc;
        s2 = fmaf(acc, acc, s2);
    }
    __shared__ float lds[WPB * 64];
    lds[wave * 64 + lane]      = s1;
    lds[wave * 64 + 32 + lane] = s2;
    __syncthreads();
    if (threadIdx.x < 64) {
        float t = 0.f;
#pragma unroll
        for (int w = 0; w < WPB; ++w) t += lds[w * 64 + threadIdx.x];
        partials[(size_t)blockIdx.x * 64 + threadIdx.x] = t;
    }
}

// ---- BN stats: 512x64 column reduce on the matrix engine -----------------
// D = ones(16x4) x B(4x16) + C  => every row of D = column sums of B.
// 4 waves x 16 columns cover the 64 stat columns; 128 WMMAs per wave.

__global__ __launch_bounds__(128)
void k_stats(const float* __restrict__ partials, const float* __restrict__ gamma,
             const float* __restrict__ beta, float* __restrict__ ss) {
    const int lane    = threadIdx.x & 31;
    const int wave    = threadIdx.x >> 5;
    const int colBase = wave * 16;
    const int nlo     = lane & 15;
    const int rofs    = (lane >= 16) ? 2 : 0;    // B layout: v0 = rows K=0/K=2, v1 = rows K=1/K=3
    v8f C = {};
    v2f A = {1.f, 1.f};                          // ones matrix (layout-invariant)
    for (int r = 0; r < AGG_BLOCKS; r += 4) {
        v2f B;
        B.x = partials[(size_t)(r + rofs)     * 64 + colBase + nlo];
        B.y = partials[(size_t)(r + rofs + 1) * 64 + colBase + nlo];
        C = __builtin_amdgcn_wmma_f32_16x16x4_f32(false, A, false, B,
                                                  (short)0, C, false, false);
    }
    __shared__ float st[64];
    if (lane < 16) st[colBase + nlo] = C[0];     // D row M=0, lanes 0..15
    __syncthreads();
    if (threadIdx.x < DIM) {
        int f = threadIdx.x;
        float mean = st[f] * (1.0f / N_NODES);
        float var  = st[32 + f] * (1.0f / N_NODES) - mean * mean;   // biased, torch-style
        float sc   = rsqrtf(var + BN_EPS) * gamma[f];
        ss[f]      = sc;
        ss[32 + f] = fmaf(-mean, sc, beta[f]);   // shift = beta - mean*scale
    }
}

// ---- affine apply: out = z*scale[f] + shift[f] ---------------------------

__global__ void k_apply(const float* __restrict__ zin, float* __restrict__ zout,
                        const float* __restrict__ ss) {
    const int total = N_NODES * DIM / 4;
    int i = blockIdx.x * blockDim.x + threadIdx.x;
    if (i >= total) return;
    float4 v = ((const float4*)zin)[i];
    int f0 = (i & 7) * 4;
    float4 o;
    o.x = fmaf(v.x, ss[f0 + 0], ss[32 + f0 + 0]);
    o.y = fmaf(v.y, ss[f0 + 1], ss[32 + f0 + 1]);
    o.z = fmaf(v.z, ss[f0 + 2], ss[32 + f0 + 2]);
    o.w = fmaf(v.w, ss[f0 + 3], ss[32 + f0 + 3]);
    ((float4*)zout)[i] = o;
}

// ---- host orchestration --------------------------------------------------

extern "C" void kernel_launch(void* const* d_in, const int* in_sizes, int n_in,
                              void* d_out, int out_size, void* d_ws, size_t ws_size,
                              hipStream_t stream) {
    const float* x      = (const float*)d_in[0];
    const int*   ei     = (const int*)d_in[1];
    const int*   e_src  = ei;
    const int*   e_dst  = ei + N_EDGES;
    const float* gammas = (const float*)d_in[2];
    const float* betas  = (const float*)d_in[3];
    float*       out    = (float*)d_out;

    char* ws = (char*)d_ws;
    size_t off = 0;
    auto carve = [&](size_t bytes) -> void* {
        void* p = ws + off;
        off = (off + bytes + 255) & ~(size_t)255;
        return p;
    };
    int*   cnt      = (int*)  carve(sizeof(int)   * N_NODES);
    float* dinv     = (float*)carve(sizeof(float) * N_NODES);
    int*   rowptr   = (int*)  carve(sizeof(int)   * (N_NODES + 1));
    int*   cursor   = (int*)  carve(sizeof(int)   * N_NODES);
    int2*  csr      = (int2*) carve(sizeof(int2)  * E_TOT);
    float* zA       = (float*)carve(sizeof(float) * N_NODES * DIM);
    float* zB       = (float*)carve(sizeof(float) * N_NODES * DIM);
    float* partials = (float*)carve(sizeof(float) * AGG_BLOCKS * 64);
    float* ss       = (float*)carve(sizeof(float) * 64);

    const int gN = (N_NODES + BLOCK - 1) / BLOCK;
    const int gE = (N_EDGES + BLOCK - 1) / BLOCK;
    const int gT = (E_TOT   + BLOCK - 1) / BLOCK;
    const int gApply = (N_NODES * DIM / 4 + BLOCK - 1) / BLOCK;

    // normalization + CSR build (per call, deterministic)
    k_init_cnt<<<gN, BLOCK, 0, stream>>>(cnt);
    k_count  <<<gE, BLOCK, 0, stream>>>(e_dst, cnt);
    k_dinv   <<<gN, BLOCK, 0, stream>>>(cnt, dinv);
    k_scan   <<<1, 1024, 0, stream>>>(cnt, rowptr, cursor);
    k_fill   <<<gT, BLOCK, 0, stream>>>(e_src, e_dst, dinv, cursor, csr);

    // 3 layers: gather+relu+stats -> wmma stat reduce -> affine apply
    const float* zin = x;
    for (int l = 0; l < N_LAYERS; ++l) {
        float* zpre  = zA;                       // relu'd aggregation
        float* znext = (l == N_LAYERS - 1) ? out : zB;
        k_agg  <<<AGG_BLOCKS, BLOCK, 0, stream>>>(zin, zpre, rowptr, csr, partials);
        k_stats<<<1, 128, 0, stream>>>(partials, gammas + l * DIM, betas + l * DIM, ss);
        k_apply<<<gApply, BLOCK, 0, stream>>>(zpre, znext, ss);
        zin = znext;
    }
}